// FilterInterpolationModule_68891275428430
// MI455X (gfx1250) — compile-verified
//
#include <hip/hip_runtime.h>
#include <cstdint>

// Fixed problem shape from the reference: B=2, C=3, H=720, W=1280, fs=4.
#define B_N   2
#define C_N   3
#define H_N   720
#define W_N   1280
#define HW_N  (H_N * W_N)          // 921600, multiple of 256
#define NPIX  (B_N * HW_N)         // 1843200 = 7200 * 256
#define TILE  256

// 16-byte vector with 4-byte alignment: emits global_load_b128 on dword-aligned
// addresses (legal under the DWORD/UNALIGNED SH_MEM alignment modes ROCm uses).
typedef float f4 __attribute__((ext_vector_type(4), aligned(4)));

__global__ __launch_bounds__(TILE)
void dain_filter_interp_kernel(const float* __restrict__ in1,   // [B,3,H,W] image
                               const float* __restrict__ in2,   // [B,2,H,W] flow (x,y)
                               const float* __restrict__ in3,   // [B,16,H,W] filter weights
                               float* __restrict__ out)         // [B,3,H,W]
{
    __shared__ float sw[16 * TILE];   // 16 KB: filter weights, [plane][pixel-in-tile]

    const int t     = threadIdx.x;
    const int base  = blockIdx.x * TILE;     // tile start (global pixel index)
    const int b     = base / HW_N;           // uniform per block (HW_N % 256 == 0)
    const int rbase = base - b * HW_N;       // tile start within batch

    // ---- CDNA5 async copy: stage this tile's 16 weight planes into LDS.
    // Wave w copies planes 2w and 2w+1; each lane moves 16 B (4 pixels).
    // input3 is streamed exactly once -> non-temporal hint keeps L2 for input1.
    {
        const int wave = t >> 5;
        const int lane = t & 31;
        const float* gplane = in3 + (size_t)(b * 16) * HW_N + rbase + lane * 4;
#pragma unroll
        for (int i = 0; i < 2; ++i) {
            const int p = wave * 2 + i;
            const float* ga  = gplane + (size_t)p * HW_N;
            uint32_t lds_off = (uint32_t)(uintptr_t)(&sw[p * TILE + lane * 4]);
            uint64_t gaddr   = (uint64_t)(uintptr_t)ga;
            asm volatile("global_load_async_to_lds_b128 %0, %1, off th:TH_LOAD_NT"
                         :: "v"(lds_off), "v"(gaddr) : "memory");
        }
    }

    // ---- per-pixel warp math (overlaps with the async copies)
    const int pix = rbase + t;
    const int y   = pix / W_N;
    const int x   = pix - y * W_N;

    const float fx = (float)x + in2[(size_t)(b * 2 + 0) * HW_N + pix];
    const float fy = (float)y + in2[(size_t)(b * 2 + 1) * HW_N + pix];
    const bool valid = (fx >= 0.0f) && (fx <= (float)(W_N - 1)) &&
                       (fy >= 0.0f) && (fy <= (float)(H_N - 1));

    const int ixi = (int)floorf(fx);
    const int iyi = (int)floorf(fy);
    const float alpha = fx - (float)ixi;
    const float beta  = fy - (float)iyi;
    const int bx = ixi - 1;                  // top-left of 4x4 window
    const int by = iyi - 1;

    const float wTL = (1.0f - alpha) * (1.0f - beta);
    const float wTR = alpha * (1.0f - beta);
    const float wBL = (1.0f - alpha) * beta;
    const float wBR = alpha * beta;

    // Clamped row slots (needed unconditionally for the row addresses).
    int rows[5];
#pragma unroll
    for (int k = 0; k < 5; ++k) {
        int r = by + k; rows[k] = (r < 0) ? 0 : ((r > H_N - 1) ? H_N - 1 : r);
    }
    // Only place the "slot k+1" pattern breaks (for valid pixels, bx,by >= -1):
    // bx == -1 -> right neighbor of dx=0 is slot 2; same for by/rows.
    const bool bxm1 = (bx == -1);
    const bool bym1 = (by == -1);

    // All needed columns always live in the 5-wide window starting at cs.
    const int  cs      = (bx < 0) ? 0 : ((bx > W_N - 5) ? (W_N - 5) : bx);
    const bool colfast = (bx == cs);         // window is exactly cols bx..bx+4

    // ---- issue ALL 30 patch loads up-front (overlap with async staging):
    // per row: one b128 + one b32, always in-bounds, branch-independent.
    float P[3][5][5];
    const float* img = in1 + (size_t)(b * 3) * HW_N;
#pragma unroll
    for (int c = 0; c < 3; ++c) {
        const float* ch = img + (size_t)c * HW_N;
#pragma unroll
        for (int r = 0; r < 5; ++r) {
            const float* rowp = ch + (size_t)rows[r] * W_N;
            const f4 v = *reinterpret_cast<const f4*>(rowp + cs);
            P[c][r][0] = v.x; P[c][r][1] = v.y; P[c][r][2] = v.z; P[c][r][3] = v.w;
            P[c][r][4] = rowp[cs + 4];
        }
    }

    // ---- single edge-remap branch (interior waves skip with one execz branch)
    if (!colfast) {
        int dk[5];
#pragma unroll
        for (int k = 0; k < 5; ++k) {
            int c = bx + k; c = (c < 0) ? 0 : ((c > W_N - 1) ? W_N - 1 : c);
            dk[k] = c - cs;                  // window slot of needed column, in [0,4]
        }
#pragma unroll
        for (int c = 0; c < 3; ++c) {
#pragma unroll
            for (int r = 0; r < 5; ++r) {
                const float t0 = P[c][r][0], t1 = P[c][r][1], t2 = P[c][r][2],
                            t3 = P[c][r][3], t4 = P[c][r][4];
#pragma unroll
                for (int k = 0; k < 5; ++k) {
                    float pv = t0;
                    pv = (dk[k] == 1) ? t1 : pv;
                    pv = (dk[k] == 2) ? t2 : pv;
                    pv = (dk[k] == 3) ? t3 : pv;
                    pv = (dk[k] == 4) ? t4 : pv;
                    P[c][r][k] = pv;
                }
            }
        }
    }

    // ---- join the async tile and make it visible to all waves
    asm volatile("s_wait_asynccnt 0" ::: "memory");
    __syncthreads();

    float w3[16];
#pragma unroll
    for (int k = 0; k < 16; ++k) w3[k] = sw[k * TILE + t];   // stride 256 -> no bank conflicts

    float acc[3];
#pragma unroll
    for (int c = 0; c < 3; ++c) {
        // bottom row for dy==0 (handles by == -1 clamp subtlety); pure selects,
        // all register indexing is compile-time constant (no scratch spills).
        float Q[5];
#pragma unroll
        for (int k = 0; k < 5; ++k) Q[k] = bym1 ? P[c][2][k] : P[c][1][k];

        float a = 0.0f;
#pragma unroll
        for (int dy = 0; dy < 4; ++dy) {
            const float T0 = P[c][dy][0], T1 = P[c][dy][1], T2 = P[c][dy][2],
                        T3 = P[c][dy][3], T4 = P[c][dy][4];
            float B0, B1, B2, B3, B4;
            if (dy == 0) { B0 = Q[0]; B1 = Q[1]; B2 = Q[2]; B3 = Q[3]; B4 = Q[4]; }
            else         { B0 = P[c][dy + 1][0]; B1 = P[c][dy + 1][1]; B2 = P[c][dy + 1][2];
                           B3 = P[c][dy + 1][3]; B4 = P[c][dy + 1][4]; }
            const float TRa = bxm1 ? T2 : T1;   // right neighbor for dx==0
            const float BRa = bxm1 ? B2 : B1;
            a = fmaf(w3[dy * 4 + 0], wTL * T0 + wTR * TRa + wBL * B0 + wBR * BRa, a);
            a = fmaf(w3[dy * 4 + 1], wTL * T1 + wTR * T2  + wBL * B1 + wBR * B2,  a);
            a = fmaf(w3[dy * 4 + 2], wTL * T2 + wTR * T3  + wBL * B2 + wBR * B3,  a);
            a = fmaf(w3[dy * 4 + 3], wTL * T3 + wTR * T4  + wBL * B3 + wBR * B4,  a);
        }
        acc[c] = valid ? a : 0.0f;
    }

    // Output is write-once -> non-temporal stores, keep L2 for input1.
#pragma unroll
    for (int c = 0; c < 3; ++c)
        __builtin_nontemporal_store(acc[c], &out[(size_t)(b * 3 + c) * HW_N + pix]);
}

extern "C" void kernel_launch(void* const* d_in, const int* in_sizes, int n_in,
                              void* d_out, int out_size, void* d_ws, size_t ws_size,
                              hipStream_t stream) {
    (void)in_sizes; (void)n_in; (void)d_ws; (void)ws_size; (void)out_size;
    const float* in1 = (const float*)d_in[0];   // [B,3,H,W] fp32
    const float* in2 = (const float*)d_in[1];   // [B,2,H,W] fp32
    const float* in3 = (const float*)d_in[2];   // [B,16,H,W] fp32
    float* out = (float*)d_out;                 // [B,3,H,W] fp32

    dim3 grid(NPIX / TILE);
    dim3 block(TILE);
    hipLaunchKernelGGL(dain_filter_interp_kernel, grid, block, 0, stream,
                       in1, in2, in3, out);
}